// spike_rnn_lif_49082886259318
// MI455X (gfx1250) — compile-verified
//
#include <hip/hip_runtime.h>

typedef __attribute__((ext_vector_type(2))) float v2f;
typedef __attribute__((ext_vector_type(8))) float v8f;
typedef __attribute__((ext_vector_type(4))) int   v4i;

#define B_ROWS 32768
#define DIN    256
#define DOUT   512
#define THRESH 0.5f

#define KCH    32              // K-chunk staged in LDS
#define LDSTR  36              // LDS row stride in floats (144B = 9*16B, conflict-free)
#define ABUF   (128 * LDSTR)   // floats per A (or B) buffer
#define NCHUNK (DIN / KCH + DOUT / KCH)   // 8 + 16 = 24

#if __has_builtin(__builtin_amdgcn_global_load_async_to_lds_b128) && \
    __has_builtin(__builtin_amdgcn_s_wait_asynccnt)
#define USE_ASYNC 1
#else
#define USE_ASYNC 0
#endif

typedef __attribute__((address_space(1))) v4i* g128_t;   // global 16B packet
typedef __attribute__((address_space(3))) v4i* l128_t;   // LDS 16B packet

// 256 threads = 8 wave32. Wave grid 2(M) x 4(N); wave tile 64x32; block tile 128x128.
__global__ __launch_bounds__(256, 2)
void snn_lif_wmma_f32(const float* __restrict__ in_spk,   // [B, DIN]
                      const float* __restrict__ mem,      // [B, DOUT]
                      const float* __restrict__ spk,      // [B, DOUT]
                      const float* __restrict__ Wd,       // [DOUT, DIN]
                      const float* __restrict__ bd,       // [DOUT]
                      const float* __restrict__ Wr,       // [DOUT, DOUT]
                      const float* __restrict__ br,       // [DOUT]
                      const float* __restrict__ tau,      // [DOUT]
                      float* __restrict__ out_mem,        // [B, DOUT]
                      float* __restrict__ out_spk)        // [B, DOUT]
{
    __shared__ float lds[2][2 * ABUF];   // [buf][A(128x36) | B(128x36)]

    const int tid  = threadIdx.x;
    const int lane = tid & 31;
    const int wave = tid >> 5;
    const int lm   = lane & 15;   // row/col within 16-wide half
    const int hi   = lane >> 4;   // selects K-pair (k+2*hi, k+2*hi+1)

    const int m_blk = blockIdx.y * 128;
    const int n_blk = blockIdx.x * 128;
    const int wM = wave & 1;      // 0..1 -> 64-row sub-tile
    const int wN = wave >> 1;     // 0..3 -> 32-col sub-tile

    v8f acc[4][2];
    #pragma unroll
    for (int i = 0; i < 4; ++i)
        #pragma unroll
        for (int j = 0; j < 2; ++j) acc[i][j] = (v8f){};

    // chunk c -> global sources (uniform per block)
    auto srcs = [&](int c, const float*& As, const float*& Bs, int& stride, int& k0) {
        if (c < DIN / KCH) {
            As = in_spk + (size_t)m_blk * DIN;
            Bs = Wd     + (size_t)n_blk * DIN;
            stride = DIN;  k0 = c * KCH;
        } else {
            As = spk + (size_t)m_blk * DOUT;
            Bs = Wr  + (size_t)n_blk * DOUT;
            stride = DOUT; k0 = (c - DIN / KCH) * KCH;
        }
    };

    auto compute = [&](int buf) {  // 64 WMMAs over one 32-deep K-chunk
        const float* Ab = &lds[buf][0];
        const float* Bb = &lds[buf][ABUF];
        #pragma unroll
        for (int ks = 0; ks < KCH; ks += 4) {
            const int kk = ks + 2 * hi;
            v2f a[4], b[2];
            #pragma unroll
            for (int i = 0; i < 4; ++i)
                a[i] = *(const v2f*)(Ab + (wM * 64 + 16 * i + lm) * LDSTR + kk);
            #pragma unroll
            for (int j = 0; j < 2; ++j)
                b[j] = *(const v2f*)(Bb + (wN * 32 + 16 * j + lm) * LDSTR + kk);
            #pragma unroll
            for (int i = 0; i < 4; ++i)
                #pragma unroll
                for (int j = 0; j < 2; ++j)
                    acc[i][j] = __builtin_amdgcn_wmma_f32_16x16x4_f32(
                        false, a[i], false, b[j], (short)0, acc[i][j], false, false);
        }
    };

#if USE_ASYNC
    // ---------- async global->LDS pipeline (no staging VGPRs) ----------
    auto issue = [&](int c, int buf) {
        const float *As, *Bs; int stride, k0;
        srcs(c, As, Bs, stride, k0);
        float* Ab = &lds[buf][0];
        float* Bb = &lds[buf][ABUF];
        #pragma unroll
        for (int i = 0; i < 4; ++i) {
            const int idx = tid + i * 256;        // 0..1023 float4 slots
            const int row = idx >> 3;             // 128 rows
            const int c4  = idx & 7;              // 8 float4 per 32-float row
            __builtin_amdgcn_global_load_async_to_lds_b128(
                (g128_t)(As + (size_t)row * stride + k0 + c4 * 4),
                (l128_t)(Ab + row * LDSTR + c4 * 4), 0, 0);
            __builtin_amdgcn_global_load_async_to_lds_b128(
                (g128_t)(Bs + (size_t)row * stride + k0 + c4 * 4),
                (l128_t)(Bb + row * LDSTR + c4 * 4), 0, 0);
        }
    };

    issue(0, 0);
    __builtin_amdgcn_s_wait_asynccnt(0);
    __syncthreads();
    for (int c = 0; c < NCHUNK; ++c) {
        if (c + 1 < NCHUNK) issue(c + 1, (c + 1) & 1);  // copies in flight during WMMAs
        compute(c & 1);
        __builtin_amdgcn_s_wait_asynccnt(0);            // next chunk's copies done (this wave)
        __syncthreads();                                // ... and visible from all waves
    }
#else
    // ---------- fallback: register-staged pipeline ----------
    float4 ra[4], rb[4];
    auto stage = [&](int c) {
        const float *As, *Bs; int stride, k0;
        srcs(c, As, Bs, stride, k0);
        #pragma unroll
        for (int i = 0; i < 4; ++i) {
            const int idx = tid + i * 256;
            const int row = idx >> 3;
            const int c4  = idx & 7;
            ra[i] = *(const float4*)(As + (size_t)row * stride + k0 + c4 * 4);
            rb[i] = *(const float4*)(Bs + (size_t)row * stride + k0 + c4 * 4);
        }
    };
    auto commit = [&](int buf) {
        float* Ab = &lds[buf][0];
        float* Bb = &lds[buf][ABUF];
        #pragma unroll
        for (int i = 0; i < 4; ++i) {
            const int idx = tid + i * 256;
            const int row = idx >> 3;
            const int c4  = idx & 7;
            *(float4*)(Ab + row * LDSTR + c4 * 4) = ra[i];
            *(float4*)(Bb + row * LDSTR + c4 * 4) = rb[i];
        }
    };
    stage(0);
    commit(0);
    __syncthreads();
    for (int c = 0; c < NCHUNK; ++c) {
        if (c + 1 < NCHUNK) stage(c + 1);
        compute(c & 1);
        if (c + 1 < NCHUNK) commit((c + 1) & 1);
        __syncthreads();
    }
#endif

    // ---- LIF epilogue ----
    #pragma unroll
    for (int j = 0; j < 2; ++j) {
        const int o = n_blk + wN * 32 + 16 * j + lm;
        const float alpha = __expf(-1.0f / tau[o]);
        const float beta  = 1.0f - alpha;
        const float bias  = bd[o] + br[o];
        #pragma unroll
        for (int i = 0; i < 4; ++i) {
            const int mbase = m_blk + wM * 64 + 16 * i + 8 * hi;
            #pragma unroll
            for (int r = 0; r < 8; ++r) {
                const size_t idx = (size_t)(mbase + r) * DOUT + o;
                const float d  = acc[i][j][r] + bias;
                const float mn = mem[idx] * alpha + beta * d - THRESH * spk[idx];
                out_mem[idx] = mn;
                out_spk[idx] = (mn - THRESH > 0.0f) ? 1.0f : 0.0f;
            }
        }
    }
}

extern "C" void kernel_launch(void* const* d_in, const int* in_sizes, int n_in,
                              void* d_out, int out_size, void* d_ws, size_t ws_size,
                              hipStream_t stream) {
    const float* in_spk = (const float*)d_in[0];   // input_spike [B, DIN]
    const float* mem    = (const float*)d_in[1];   // mem         [B, DOUT]
    const float* spk    = (const float*)d_in[2];   // spike       [B, DOUT]
    const float* Wd     = (const float*)d_in[3];   // W_dense     [DOUT, DIN]
    const float* bd     = (const float*)d_in[4];   // b_dense     [DOUT]
    const float* Wr     = (const float*)d_in[5];   // W_rec       [DOUT, DOUT]
    const float* br     = (const float*)d_in[6];   // b_rec       [DOUT]
    const float* tau    = (const float*)d_in[7];   // tau_m       [DOUT]

    float* out_mem = (float*)d_out;
    float* out_spk = (float*)d_out + (size_t)B_ROWS * DOUT;

    dim3 grid(DOUT / 128, B_ROWS / 128);
    snn_lif_wmma_f32<<<grid, 256, 0, stream>>>(in_spk, mem, spk, Wd, bd, Wr, br, tau,
                                               out_mem, out_spk);
}